// DustNet_71021579207072
// MI455X (gfx1250) — compile-verified
//
#include <hip/hip_runtime.h>

// ---------------------------------------------------------------------------
// DustNet forward for MI455X (gfx1250), all GEMMs on V_WMMA_F32_16X16X4_F32.
//
// Exact-math simplifications vs the reference:
//  * only LISTA layer K-1 is computed (reference discards layers 0..K-2)
//  * everything else is computed faithfully: X, G, softmax(G^T G), Z, mat, out
//
// Codegen strategy:
//  * LDS tiles are M-major (As[m][k]) / N-major (Bs[n][k]) with row stride 20
//    words: each WMMA fragment is ONE aligned ds_load_b64 straight into an
//    even VGPR pair (no v_mov repacking), conflict-free across 64 banks.
//  * double-buffered LDS + register staging: global loads for tile t+1 are
//    in flight while 32 wmma ops run on tile t.
//
// Workspace layout (floats), total 27,262,976 floats = ~104 MiB:
//   X  [512 x4096]   G [1024x4096]   L/attn [4096x4096]
//   Z  [512 x4096]   M1 [512 x4096]  (M1 reused in-place for thresholded H)
// ---------------------------------------------------------------------------

typedef __attribute__((ext_vector_type(2))) float f32x2;
typedef __attribute__((ext_vector_type(8))) float f32x8;

#define SIGMA_C 0.1f
#define MU_C    0.0f

constexpr int BM = 128, BN = 128, BK = 16;
constexpr int TS = 20;   // LDS row stride (words): mult. of 4 (b128/b64 align),
                         // 20*m mod 64 hits all 16 multiples of 4 -> b64 frag
                         // reads are bank-conflict-free.

enum { EPI_NONE = 0, EPI_ADDNOISE = 1, EPI_SCALE = 2, EPI_ACCTHRESH = 3 };

// C[M,N] = op(A) @ B  (+ epilogue).  TRANSA==0: A row-major [M,K].
// TRANSA==1: A_eff[m,k] = Astore[k*lda + m]  (used for G^T @ G).
// B row-major [K,N].  M,N multiples of 128; K multiple of 16.
template <int TRANSA, int EPI>
__global__ __launch_bounds__(256) void gemm_wmma_f32(
    const float* __restrict__ A, const float* __restrict__ B, float* C,
    int M, int N, int K, int lda, int ldb, int ldc,
    const float* aux, const float* sp0, const float* sp1)
{
  __shared__ __align__(16) float As[2][BM][TS];   // m-major: As[.][m][k]
  __shared__ __align__(16) float Bs[2][BN][TS];   // n-major: Bs[.][n][k]

  const int tid  = threadIdx.x;
  const int lane = tid & 31;
  const int hl   = lane >> 4;        // half-wave (0/1)
  const int lrow = lane & 15;
  const int w    = tid >> 5;         // wave id 0..7
  const int wmBase = (w >> 1) * 32;  // 4 waves along M (32 rows each)
  const int wnBase = (w & 1) * 64;   // 2 waves along N (64 cols each)

  const int bm0 = blockIdx.y * BM;
  const int bn0 = blockIdx.x * BN;

  // staging index maps (per thread)
  const int rA = (tid + 0) >> 2, cA0 = ((tid + 0) & 3) << 2;   // TRANSA==0, part 0
  const int rB = (tid + 256) >> 2, cB0 = ((tid + 256) & 3) << 2; // part 1 (rB=rA+64)
  const int mT = tid & 127;                                    // TRANSA==1 / B
  const int cT0 = (tid >> 7) << 2;                             // 0 or 4 (part0), +8 (part1)

  f32x8 acc[2][4] = {};   // wave tile 32x64 = 2x4 grid of 16x16 f32 accumulators

  float4 ra0, ra1, rb0, rb1;   // register staging for the next tile

  auto stage = [&](int kt) {
    if (TRANSA == 0) {
      ra0 = *reinterpret_cast<const float4*>(A + (size_t)(bm0 + rA) * lda + kt + cA0);
      ra1 = *reinterpret_cast<const float4*>(A + (size_t)(bm0 + rB) * lda + kt + cB0);
    } else {
      const float* p0 = A + (size_t)(kt + cT0) * lda + bm0 + mT;
      ra0 = make_float4(p0[0], p0[lda], p0[2 * (size_t)lda], p0[3 * (size_t)lda]);
      const float* p1 = p0 + 8 * (size_t)lda;
      ra1 = make_float4(p1[0], p1[lda], p1[2 * (size_t)lda], p1[3 * (size_t)lda]);
    }
    {
      const float* q0 = B + (size_t)(kt + cT0) * ldb + bn0 + mT;
      rb0 = make_float4(q0[0], q0[ldb], q0[2 * (size_t)ldb], q0[3 * (size_t)ldb]);
      const float* q1 = q0 + 8 * (size_t)ldb;
      rb1 = make_float4(q1[0], q1[ldb], q1[2 * (size_t)ldb], q1[3 * (size_t)ldb]);
    }
  };

  auto commit = [&](int p) {
    if (TRANSA == 0) {
      *reinterpret_cast<float4*>(&As[p][rA][cA0]) = ra0;
      *reinterpret_cast<float4*>(&As[p][rB][cB0]) = ra1;
    } else {
      *reinterpret_cast<float4*>(&As[p][mT][cT0]) = ra0;
      *reinterpret_cast<float4*>(&As[p][mT][cT0 + 8]) = ra1;
    }
    *reinterpret_cast<float4*>(&Bs[p][mT][cT0]) = rb0;
    *reinterpret_cast<float4*>(&Bs[p][mT][cT0 + 8]) = rb1;
  };

  auto compute = [&](int p) {
    #pragma unroll
    for (int kk = 0; kk < BK; kk += 4) {
      // ISA 7.12.2, 32-bit operands: fragment element j = [row, 2*half + j]
      // -> one b64 LDS load per fragment, already even-pair aligned for WMMA.
      f32x2 af[2], bf[4];
      #pragma unroll
      for (int i = 0; i < 2; ++i)
        af[i] = *reinterpret_cast<const f32x2*>(
            &As[p][wmBase + i * 16 + lrow][kk + 2 * hl]);
      #pragma unroll
      for (int n = 0; n < 4; ++n)
        bf[n] = *reinterpret_cast<const f32x2*>(
            &Bs[p][wnBase + n * 16 + lrow][kk + 2 * hl]);
      #pragma unroll
      for (int i = 0; i < 2; ++i)
        #pragma unroll
        for (int n = 0; n < 4; ++n)
          acc[i][n] = __builtin_amdgcn_wmma_f32_16x16x4_f32(
              false, af[i], false, bf[n], (short)0, acc[i][n], false, false);
    }
  };

  // software pipeline: stage(t+1) overlaps compute(t)
  stage(0);
  commit(0);
  int p = 0;
  for (int kt = 0; kt < K; kt += BK) {
    __syncthreads();                   // buffer p staged; buffer 1-p free
    const bool more = (kt + BK) < K;
    if (more) stage(kt + BK);          // global loads in flight ...
    compute(p);                        // ... behind 32 wmma ops
    if (more) commit(1 - p);           // drain into the other buffer
    p ^= 1;
  }

  // ---- epilogue ----
  float thr = 0.f, scale = 1.f;
  if (EPI == EPI_SCALE)     scale = sp0[0];
  if (EPI == EPI_ACCTHRESH) thr   = sp0[0] / sp1[0];

  #pragma unroll
  for (int i = 0; i < 2; ++i)
    #pragma unroll
    for (int n = 0; n < 4; ++n)
      #pragma unroll
      for (int v = 0; v < 8; ++v) {
        // C/D layout: VGPR v holds row (v + 8*half), col = lane%16
        int row = bm0 + wmBase + i * 16 + v + 8 * hl;
        int col = bn0 + wnBase + n * 16 + lrow;
        size_t idx = (size_t)row * ldc + col;
        float val = acc[i][n][v];
        if (EPI == EPI_ADDNOISE) {
          val += SIGMA_C * aux[idx] + MU_C;
        } else if (EPI == EPI_SCALE) {
          val *= scale;
        } else if (EPI == EPI_ACCTHRESH) {
          float t = val + aux[idx];     // aux aliases C (in-place): RMW per element
          float a = fabsf(t) - thr;
          val = (a > 0.f) ? copysignf(a, t) : 0.f;
        }
        C[idx] = val;
      }
}

// Row softmax over a [4096 x 4096] matrix, in place. One 256-thread block per
// row; 16 elements per thread held in registers (single global read + write).
__global__ __launch_bounds__(256) void softmax_rows_4096(float* P)
{
  const int T = 4096;
  float* row = P + (size_t)blockIdx.x * T;
  const int tid = threadIdx.x;
  const int wid = tid >> 5;

  float v[16];
  float m = -__builtin_inff();
  #pragma unroll
  for (int i = 0; i < 16; ++i) { v[i] = row[tid + i * 256]; m = fmaxf(m, v[i]); }

  #pragma unroll
  for (int off = 16; off >= 1; off >>= 1) m = fmaxf(m, __shfl_xor(m, off, 32));
  __shared__ float redm[8];
  if ((tid & 31) == 0) redm[wid] = m;
  __syncthreads();
  float mrow = redm[0];
  #pragma unroll
  for (int q = 1; q < 8; ++q) mrow = fmaxf(mrow, redm[q]);

  float s = 0.f;
  #pragma unroll
  for (int i = 0; i < 16; ++i) { v[i] = __expf(v[i] - mrow); s += v[i]; }
  #pragma unroll
  for (int off = 16; off >= 1; off >>= 1) s += __shfl_xor(s, off, 32);
  __shared__ float reds[8];
  if ((tid & 31) == 0) reds[wid] = s;
  __syncthreads();
  float srow = 0.f;
  #pragma unroll
  for (int q = 0; q < 8; ++q) srow += reds[q];

  const float inv = 1.f / srow;
  #pragma unroll
  for (int i = 0; i < 16; ++i) row[tid + i * 256] = v[i] * inv;
}

extern "C" void kernel_launch(void* const* d_in, const int* in_sizes, int n_in,
                              void* d_out, int out_size, void* d_ws, size_t ws_size,
                              hipStream_t stream) {
  (void)in_sizes; (void)n_in; (void)out_size; (void)ws_size;
  const float* S  = (const float*)d_in[0];   // [1024,4096]
  const float* E  = (const float*)d_in[1];   // [512,4096]
  const float* A  = (const float*)d_in[2];   // [512,1024]
  const float* Dm = (const float*)d_in[3];   // [1024,512]
  const float* U  = (const float*)d_in[4];   // [8,512,512]
  const float* V  = (const float*)d_in[5];   // [8,512,512]
  const float* l1 = (const float*)d_in[6];
  const float* l2 = (const float*)d_in[7];
  const float* c  = (const float*)d_in[8];
  const float* H0 = (const float*)d_in[9];   // [512,4096]
  float* out = (float*)d_out;                // [1024,4096]

  const int Mm = 512, Nn = 1024, Dd = 512, T = 4096, Kl = 8;

  float* ws = (float*)d_ws;
  float* X  = ws;                         // 512*4096
  float* G  = X  + (size_t)Mm * T;        // 1024*4096
  float* L  = G  + (size_t)Nn * T;        // 4096*4096
  float* Z  = L  + (size_t)T  * T;        // 512*4096
  float* M1 = Z  + (size_t)Dd * T;        // 512*4096   (mat -> H in place)

  const float* U7 = U + (size_t)(Kl - 1) * Dd * Dd;
  const float* V7 = V + (size_t)(Kl - 1) * Dd * Mm;

  dim3 blk(256);

  // 1) X = A@S + SIGMA*E + MU                         [512,4096], K=1024
  gemm_wmma_f32<0, EPI_ADDNOISE><<<dim3(T / 128, Mm / 128), blk, 0, stream>>>(
      A, S, X, Mm, T, Nn, Nn, T, T, E, nullptr, nullptr);
  // 2) G = D@H0                                       [1024,4096], K=512
  gemm_wmma_f32<0, EPI_NONE><<<dim3(T / 128, Nn / 128), blk, 0, stream>>>(
      Dm, H0, G, Nn, T, Dd, Dd, T, T, nullptr, nullptr, nullptr);
  // 3) L = G^T @ G   (attention logits)               [4096,4096], K=1024
  gemm_wmma_f32<1, EPI_NONE><<<dim3(T / 128, T / 128), blk, 0, stream>>>(
      G, G, L, T, T, Nn, T, T, T, nullptr, nullptr, nullptr);
  // 4) attn = softmax(L, axis=-1), in place
  softmax_rows_4096<<<dim3(T), blk, 0, stream>>>(L);
  // 5) Z = l2 * (H0 @ attn)                           [512,4096], K=4096
  gemm_wmma_f32<0, EPI_SCALE><<<dim3(T / 128, Dd / 128), blk, 0, stream>>>(
      H0, L, Z, Dd, T, T, T, T, T, nullptr, l2, nullptr);
  // 6) M1 = U[K-1] @ Z                                [512,4096], K=512
  gemm_wmma_f32<0, EPI_NONE><<<dim3(T / 128, Dd / 128), blk, 0, stream>>>(
      U7, Z, M1, Dd, T, Dd, Dd, T, T, nullptr, nullptr, nullptr);
  // 7) M1 = softthresh(V[K-1]@X + M1, l1/c)  (in place)
  gemm_wmma_f32<0, EPI_ACCTHRESH><<<dim3(T / 128, Dd / 128), blk, 0, stream>>>(
      V7, X, M1, Dd, T, Mm, Mm, T, T, M1, l1, c);
  // 8) out = D @ H_last                               [1024,4096], K=512
  gemm_wmma_f32<0, EPI_NONE><<<dim3(T / 128, Nn / 128), blk, 0, stream>>>(
      Dm, M1, out, Nn, T, Dd, Dd, T, T, nullptr, nullptr, nullptr);
}